// GAT_full_57578331570300
// MI455X (gfx1250) — compile-verified
//
#include <hip/hip_runtime.h>

#define N_NODES   100000
#define H_HEADS   4
#define C_FEATS   32
#define E_EDGES   1600000
#define IN_FEATS  256
#define HC        128          // H*C
#define NEG_SLOPE 0.2f

typedef __attribute__((ext_vector_type(2))) float v2f;
typedef __attribute__((ext_vector_type(8))) float v8f;

// ---------------------------------------------------------------------------
// Kernel 1: feat = x @ W  via V_WMMA_F32_16X16X4_F32.
// Block = 256 threads = 8 waves; block computes a 16x128 output tile,
// wave w computes the 16x16 sub-tile at columns [16w, 16w+16).
// ---------------------------------------------------------------------------
__global__ __launch_bounds__(256) void gat_gemm_wmma(
    const float* __restrict__ x, const float* __restrict__ W,
    float* __restrict__ feat)
{
  // 16 rows x 256 cols of x, row stride padded to 260 (260 % 64 == 4 banks)
  __shared__ float sx[16 * 260];

  const int rowBase = blockIdx.x * 16;       // N is an exact multiple of 16
  const int t = threadIdx.x;

  // Cooperative coalesced load of the A tile: thread t loads column t of all rows
  #pragma unroll
  for (int i = 0; i < 16; ++i)
    sx[i * 260 + t] = x[(size_t)(rowBase + i) * IN_FEATS + t];
  __syncthreads();

  const int wave    = t >> 5;                // 0..7
  const int lane    = t & 31;
  const int colBase = wave * 16;
  const int mn      = lane & 15;             // A row (M) == B col (N) index
  const int koff    = (lane >> 4) << 1;      // lanes 0-15: K+0/K+1, lanes 16-31: K+2/K+3

  v8f acc = {};
  #pragma unroll 4
  for (int k0 = 0; k0 < IN_FEATS; k0 += 4) {
    v2f a, b;
    a.x = sx[mn * 260 + k0 + koff];
    a.y = sx[mn * 260 + k0 + koff + 1];
    b.x = W[(size_t)(k0 + koff)     * HC + colBase + mn];
    b.y = W[(size_t)(k0 + koff + 1) * HC + colBase + mn];
    // D = A(16x4) * B(4x16) + C ; args: neg_a, A, neg_b, B, c_mod, C, reuse_a, reuse_b
    acc = __builtin_amdgcn_wmma_f32_16x16x4_f32(false, a, false, b,
                                                (short)0, acc, false, false);
  }

  // C/D layout: VGPR r -> row r (lanes 0-15) or row r+8 (lanes 16-31)
  const int rOff = (lane < 16) ? 0 : 8;
  const int col  = colBase + mn;
  #pragma unroll
  for (int r = 0; r < 8; ++r)
    feat[(size_t)(rowBase + rOff + r) * HC + col] = acc[r];
}

// ---------------------------------------------------------------------------
// Kernel 2: el[n,h] = <feat[n,h,:], attn_l[h,:]> ; er likewise.
// One thread per (n,h); idx*C == n*HC + h*C so feat reads are contiguous.
// ---------------------------------------------------------------------------
__global__ void gat_attn(const float* __restrict__ feat,
                         const float* __restrict__ attn_l,
                         const float* __restrict__ attn_r,
                         float* __restrict__ el, float* __restrict__ er)
{
  int idx = blockIdx.x * blockDim.x + threadIdx.x;   // n*H + h
  if (idx >= N_NODES * H_HEADS) return;
  int h = idx & (H_HEADS - 1);
  const float4* f4 = (const float4*)(feat + (size_t)idx * C_FEATS);
  const float4* a4 = (const float4*)(attn_l + h * C_FEATS);
  const float4* b4 = (const float4*)(attn_r + h * C_FEATS);
  float sl = 0.f, sr = 0.f;
  #pragma unroll
  for (int i = 0; i < C_FEATS / 4; ++i) {
    float4 f = f4[i], a = a4[i], b = b4[i];
    sl += f.x * a.x + f.y * a.y + f.z * a.z + f.w * a.w;
    sr += f.x * b.x + f.y * b.y + f.z * b.z + f.w * b.w;
  }
  el[idx] = sl;
  er[idx] = sr;
}

// Monotonic-int float atomic max (works when dst initialized to -inf).
__device__ inline void atomic_max_f32(float* addr, float val) {
  if (val >= 0.0f)
    atomicMax((int*)addr, __float_as_int(val));
  else
    atomicMin((unsigned int*)addr, __float_as_uint(val));
}

__device__ inline float leaky(float v) { return v > 0.f ? v : NEG_SLOPE * v; }

__global__ void fill_neginf(float* __restrict__ p, int n) {
  int i = blockIdx.x * blockDim.x + threadIdx.x;
  if (i < n) p[i] = -__builtin_inff();
}

// ---------------------------------------------------------------------------
// Kernel 3: per-edge segment max  m[dst,h] = max leaky(el[src,h]+er[dst,h])
// ---------------------------------------------------------------------------
__global__ void gat_edge_max(const int* __restrict__ src, const int* __restrict__ dst,
                             const float* __restrict__ el, const float* __restrict__ er,
                             float* __restrict__ m)
{
  int e = blockIdx.x * blockDim.x + threadIdx.x;
  if (e >= E_EDGES) return;
  int s = src[e], d = dst[e];
  float4 l4 = *(const float4*)(el + (size_t)s * H_HEADS);
  float4 r4 = *(const float4*)(er + (size_t)d * H_HEADS);
  float* md = m + (size_t)d * H_HEADS;
  atomic_max_f32(md + 0, leaky(l4.x + r4.x));
  atomic_max_f32(md + 1, leaky(l4.y + r4.y));
  atomic_max_f32(md + 2, leaky(l4.z + r4.z));
  atomic_max_f32(md + 3, leaky(l4.w + r4.w));
}

// ---------------------------------------------------------------------------
// Kernel 4: ex[e,h] = exp(s - m[dst,h]);  ssum[dst,h] += ex
// ---------------------------------------------------------------------------
__global__ void gat_edge_exp(const int* __restrict__ src, const int* __restrict__ dst,
                             const float* __restrict__ el, const float* __restrict__ er,
                             const float* __restrict__ m,
                             float* __restrict__ ex, float* __restrict__ ssum)
{
  int e = blockIdx.x * blockDim.x + threadIdx.x;
  if (e >= E_EDGES) return;
  int s = src[e], d = dst[e];
  float4 l4 = *(const float4*)(el + (size_t)s * H_HEADS);
  float4 r4 = *(const float4*)(er + (size_t)d * H_HEADS);
  float4 m4 = *(const float4*)(m  + (size_t)d * H_HEADS);
  float e0 = __expf(leaky(l4.x + r4.x) - m4.x);
  float e1 = __expf(leaky(l4.y + r4.y) - m4.y);
  float e2 = __expf(leaky(l4.z + r4.z) - m4.z);
  float e3 = __expf(leaky(l4.w + r4.w) - m4.w);
  *(float4*)(ex + (size_t)e * H_HEADS) = make_float4(e0, e1, e2, e3);
  float* sd = ssum + (size_t)d * H_HEADS;
  atomicAdd(sd + 0, e0);
  atomicAdd(sd + 1, e1);
  atomicAdd(sd + 2, e2);
  atomicAdd(sd + 3, e3);
}

// ---------------------------------------------------------------------------
// Kernel 5: agg[dst,:] += (ex/ssum) * feat[src,:]   (one wave per edge,
// lane l handles channels [4l,4l+4); coalesced 512B gather of feat[src])
// ---------------------------------------------------------------------------
__global__ __launch_bounds__(256) void gat_scatter(
    const int* __restrict__ src, const int* __restrict__ dst,
    const float* __restrict__ ex, const float* __restrict__ ssum,
    const float* __restrict__ feat, float* __restrict__ agg)
{
  long long gid = (long long)blockIdx.x * blockDim.x + threadIdx.x;
  int e = (int)(gid >> 5);
  if (e >= E_EDGES) return;
  int lane = (int)(gid & 31);
  int s = src[e], d = dst[e];
  int h = lane >> 3;                                   // head = (4*lane)/32
  float alpha = ex[(size_t)e * H_HEADS + h] / ssum[(size_t)d * H_HEADS + h];
  float4 f = *(const float4*)(feat + (size_t)s * HC + lane * 4);
  float* ap = agg + (size_t)d * HC + lane * 4;
  atomicAdd(ap + 0, alpha * f.x);
  atomicAdd(ap + 1, alpha * f.y);
  atomicAdd(ap + 2, alpha * f.z);
  atomicAdd(ap + 3, alpha * f.w);
}

// ---------------------------------------------------------------------------
// Kernel 6: out[n,c] = relu(mean_h(agg[n,h,c] + bias[h*C+c]))
// ---------------------------------------------------------------------------
__global__ void gat_finalize(const float* __restrict__ agg,
                             const float* __restrict__ bias,
                             float* __restrict__ out)
{
  int idx = blockIdx.x * blockDim.x + threadIdx.x;     // n*C + c
  if (idx >= N_NODES * C_FEATS) return;
  int n = idx >> 5, c = idx & 31;
  float sum = 0.f;
  #pragma unroll
  for (int h = 0; h < H_HEADS; ++h)
    sum += agg[(size_t)n * HC + h * C_FEATS + c] + bias[h * C_FEATS + c];
  float v = sum * (1.0f / H_HEADS);
  out[idx] = v > 0.f ? v : 0.f;
}

// ---------------------------------------------------------------------------
extern "C" void kernel_launch(void* const* d_in, const int* in_sizes, int n_in,
                              void* d_out, int out_size, void* d_ws, size_t ws_size,
                              hipStream_t stream)
{
  const float* x      = (const float*)d_in[0];
  const int*   src    = (const int*)  d_in[1];
  const int*   dst    = (const int*)  d_in[2];
  const float* W      = (const float*)d_in[3];
  const float* attn_l = (const float*)d_in[4];
  const float* attn_r = (const float*)d_in[5];
  const float* bias   = (const float*)d_in[6];
  float* out = (float*)d_out;

  char* ws = (char*)d_ws;
  size_t off = 0;
  auto alloc = [&](size_t bytes) -> void* {
    void* p = (void*)(ws + off);
    off += (bytes + 255) & ~((size_t)255);
    return p;
  };
  float* feat = (float*)alloc((size_t)N_NODES * HC * sizeof(float));       // 51.2 MB
  float* el   = (float*)alloc((size_t)N_NODES * H_HEADS * sizeof(float));  // 1.6 MB
  float* er   = (float*)alloc((size_t)N_NODES * H_HEADS * sizeof(float));
  float* m    = (float*)alloc((size_t)N_NODES * H_HEADS * sizeof(float));
  float* ssum = (float*)alloc((size_t)N_NODES * H_HEADS * sizeof(float));
  float* ex   = (float*)alloc((size_t)E_EDGES * H_HEADS * sizeof(float));  // 25.6 MB
  float* agg  = (float*)alloc((size_t)N_NODES * HC * sizeof(float));       // 51.2 MB

  hipMemsetAsync(ssum, 0, (size_t)N_NODES * H_HEADS * sizeof(float), stream);
  hipMemsetAsync(agg,  0, (size_t)N_NODES * HC * sizeof(float), stream);

  fill_neginf<<<(N_NODES * H_HEADS + 255) / 256, 256, 0, stream>>>(m, N_NODES * H_HEADS);

  gat_gemm_wmma<<<N_NODES / 16, 256, 0, stream>>>(x, W, feat);

  gat_attn<<<(N_NODES * H_HEADS + 255) / 256, 256, 0, stream>>>(feat, attn_l, attn_r, el, er);

  gat_edge_max<<<(E_EDGES + 255) / 256, 256, 0, stream>>>(src, dst, el, er, m);

  gat_edge_exp<<<(E_EDGES + 255) / 256, 256, 0, stream>>>(src, dst, el, er, m, ex, ssum);

  long long scatter_threads = (long long)E_EDGES * 32;
  gat_scatter<<<(unsigned)((scatter_threads + 255) / 256), 256, 0, stream>>>(
      src, dst, ex, ssum, feat, agg);

  gat_finalize<<<(N_NODES * C_FEATS + 255) / 256, 256, 0, stream>>>(agg, bias, out);
}